// ElectricOverflow_26104811225785
// MI455X (gfx1250) — compile-verified
//
#include <hip/hip_runtime.h>

// ---------------------------------------------------------------------------
// Problem constants (match the reference exactly)
// ---------------------------------------------------------------------------
#define NMOV 250000
#define NTER 8000
#define NFIL 60000
#define NTOT (NMOV + NTER + NFIL)      // 318000
#define GNX 512
#define GNY 512
#define GNZ 8
#define NBINS (GNX * GNY * GNZ)        // 2097152 bins = 8 MB fp32 (L2-resident)
#define SQRT2F 1.41421356237309515f

typedef __attribute__((ext_vector_type(2))) float v2f;
typedef __attribute__((ext_vector_type(8))) float v8f;

// ---------------------------------------------------------------------------
// Kernel 0: zero the density grid + scalar accumulators (runs every launch,
// so graph replays are deterministic).
// ---------------------------------------------------------------------------
__global__ void eo_init(float* __restrict__ grid, unsigned* __restrict__ max_slot,
                        float* __restrict__ out) {
  int i = blockIdx.x * blockDim.x + threadIdx.x;
  const int n4 = NBINS / 4;
  float4 z = make_float4(0.f, 0.f, 0.f, 0.f);
  for (int f = i; f < n4; f += gridDim.x * blockDim.x)
    reinterpret_cast<float4*>(grid)[f] = z;          // global_store_b128
  if (i == 0) {
    *max_slot = 0u;
    out[0] = 0.0f;
    out[1] = 0.0f;
  }
}

// ---------------------------------------------------------------------------
// Exact overlap-weighted scatter of one box into the grid.
// Equivalent to the reference's fixed-K clip/valid logic: every bin with
// positive overlap lies in [floor(p0), floor(p0+s)]; clamped bins outside
// [0,NB) are the reference's `valid=false` (dropped) entries.
// ---------------------------------------------------------------------------
__device__ __forceinline__ void eo_scatter_box(float* __restrict__ grid,
                                               float x0, float y0, float z0,
                                               float sx, float sy, float sz,
                                               float w) {
  int ix0 = (int)floorf(x0), ix1 = (int)floorf(x0 + sx);
  int iy0 = (int)floorf(y0), iy1 = (int)floorf(y0 + sy);
  int iz0 = (int)floorf(z0), iz1 = (int)floorf(z0 + sz);
  ix0 = max(ix0, 0); ix1 = min(ix1, GNX - 1);
  iy0 = max(iy0, 0); iy1 = min(iy1, GNY - 1);
  iz0 = max(iz0, 0); iz1 = min(iz1, GNZ - 1);
  float xe = x0 + sx, ye = y0 + sy, ze = z0 + sz;
  for (int kx = ix0; kx <= ix1; ++kx) {
    float ox = fmaxf(fminf(xe, (float)(kx + 1)) - fmaxf(x0, (float)kx), 0.0f);
    float wx = w * ox;
    for (int ky = iy0; ky <= iy1; ++ky) {
      float oy = fmaxf(fminf(ye, (float)(ky + 1)) - fmaxf(y0, (float)ky), 0.0f);
      float wxy = wx * oy;
      int base = (kx * GNY + ky) * GNZ;
      for (int kz = iz0; kz <= iz1; ++kz) {
        float oz = fmaxf(fminf(ze, (float)(kz + 1)) - fmaxf(z0, (float)kz), 0.0f);
        atomicAdd(&grid[base + kz], wxy * oz);       // global_atomic_add_f32 (L2)
      }
    }
  }
}

// ---------------------------------------------------------------------------
// Kernel 1: terminals (indices [NMOV, NMOV+NTER)), weight 1, raw sizes.
// ---------------------------------------------------------------------------
__global__ void eo_scatter_fixed(const float* __restrict__ pos,
                                 const float* __restrict__ nsx,
                                 const float* __restrict__ nsy,
                                 const float* __restrict__ nsz,
                                 float* __restrict__ grid) {
  int t = blockIdx.x * blockDim.x + threadIdx.x;
  if (t >= NTER) return;
  int i = NMOV + t;
  eo_scatter_box(grid, pos[i], pos[NTOT + i], pos[2 * NTOT + i],
                 nsx[i], nsy[i], nsz[i], 1.0f);
}

// ---------------------------------------------------------------------------
// Kernel 2: movable + fillers. Sizes clamped to sqrt(2), centered offset,
// weight = volume ratio.
// ---------------------------------------------------------------------------
__global__ void eo_scatter_mov(const float* __restrict__ pos,
                               const float* __restrict__ nsx,
                               const float* __restrict__ nsy,
                               const float* __restrict__ nsz,
                               float* __restrict__ grid) {
  int t = blockIdx.x * blockDim.x + threadIdx.x;
  if (t >= NMOV + NFIL) return;
  int i = (t < NMOV) ? t : (t - NMOV) + (NTOT - NFIL);
  float sx = nsx[i], sy = nsy[i], sz = nsz[i];
  float csx = fmaxf(sx, SQRT2F);
  float csy = fmaxf(sy, SQRT2F);
  float csz = fmaxf(sz, SQRT2F);
  float w = (sx * sy * sz) / (csx * csy * csz);
  float x0 = pos[i]            + (sx - csx) * 0.5f;
  float y0 = pos[NTOT + i]     + (sy - csy) * 0.5f;
  float z0 = pos[2 * NTOT + i] + (sz - csz) * 0.5f;
  eo_scatter_box(grid, x0, y0, z0, csx, csy, csz, w);
}

// ---------------------------------------------------------------------------
// Kernel 3: fused padding-mask + cost/max reduction.
// WMMA f32 16x16x4 with B = ones acts as a 64-elem/instr fp32 adder tree:
// D[m,n] = rowsum(m) for all n, so sum(all 8 C VGPRs over all 32 lanes)
// equals 16 * (total sum); each lane contributes laneSum/16.
// 1024 waves x 32 iters x 64 elems == NBINS exactly -> fully uniform control
// flow, EXEC all-ones through every v_wmma (ISA requirement).
// ---------------------------------------------------------------------------
__global__ void __launch_bounds__(256) eo_reduce(const float* __restrict__ grid,
                                                 float* __restrict__ out,
                                                 unsigned* __restrict__ max_slot) {
  const int lane = threadIdx.x & 31;
  const int wave = (blockIdx.x * blockDim.x + threadIdx.x) >> 5;  // 0..1023

  v8f acc = {};                 // C accumulator, 8 VGPRs
  v2f ones; ones.x = 1.0f; ones.y = 1.0f;
  float mx = 0.0f;

  #pragma unroll 4
  for (int j = 0; j < 32; ++j) {
    int f = wave * 2048 + j * 64 + lane * 2;        // even flat index
    v2f d2 = *reinterpret_cast<const v2f*>(grid + f);  // global_load_b64

    // flat -> (kx,ky,kz); NZ=8, NY=512
    int kz0 = f & 7;            // even: {0,2,4,6}; second elem is kz0+1
    int ky  = (f >> 3) & 511;
    int kx  = f >> 12;
    bool exy = (ky == 0) | (ky == GNY - 1) | (kx == 0) | (kx == GNX - 1);
    float d0 = (exy | (kz0 == 0)) ? 1.0f : d2.x;    // padding shell := TD*BIN_VOL
    float d1 = (exy | (kz0 == 6)) ? 1.0f : d2.y;    // kz0+1==7 edge

    mx = fmaxf(mx, fmaxf(d0, d1));

    v2f a;
    a.x = fmaxf(d0 - 1.0f, 0.0f);
    a.y = fmaxf(d1 - 1.0f, 0.0f);
    // v_wmma_f32_16x16x4_f32: acc += A(16x4) * ones(4x16)
    acc = __builtin_amdgcn_wmma_f32_16x16x4_f32(
        /*neg_a=*/false, a, /*neg_b=*/false, ones,
        /*c_mod=*/(short)0, acc, /*reuse_a=*/false, /*reuse_b=*/false);
  }

  float laneSum = ((acc[0] + acc[1]) + (acc[2] + acc[3])) +
                  ((acc[4] + acc[5]) + (acc[6] + acc[7]));
  laneSum *= (1.0f / 16.0f);    // undo 16-column replication (exact)

  __shared__ float    s_cost;
  __shared__ unsigned s_max;
  if (threadIdx.x == 0) { s_cost = 0.0f; s_max = 0u; }
  __syncthreads();
  atomicAdd(&s_cost, laneSum);                      // ds_add_f32
  atomicMax(&s_max, __float_as_uint(mx));           // ds_max_u32 (d>=0 monotone)
  __syncthreads();
  if (threadIdx.x == 0) {
    atomicAdd(&out[0], s_cost);                     // global_atomic_add_f32
    atomicMax(max_slot, s_max);                     // global_atomic_max_u32
  }
}

// ---------------------------------------------------------------------------
// Kernel 4: publish max_density (BIN_VOL == 1).
// ---------------------------------------------------------------------------
__global__ void eo_finalize(const unsigned* __restrict__ max_slot,
                            float* __restrict__ out) {
  if (blockIdx.x == 0 && threadIdx.x == 0)
    out[1] = __uint_as_float(*max_slot);
}

// ---------------------------------------------------------------------------
// Host launcher: all kernels sequential on `stream`; ws = [8 MB grid | u32 max]
// ---------------------------------------------------------------------------
extern "C" void kernel_launch(void* const* d_in, const int* in_sizes, int n_in,
                              void* d_out, int out_size, void* d_ws, size_t ws_size,
                              hipStream_t stream) {
  const float* pos = (const float*)d_in[0];          // 3*N
  const float* nsx = (const float*)d_in[1];          // N
  const float* nsy = (const float*)d_in[2];          // N
  const float* nsz = (const float*)d_in[3];          // N
  // d_in[4..6] bin centers and d_in[7] sorted_node_map are unused (as in ref).

  float*    out      = (float*)d_out;                // [cost, max_density]
  float*    grid     = (float*)d_ws;                 // NBINS floats
  unsigned* max_slot = (unsigned*)((char*)d_ws + (size_t)NBINS * sizeof(float));

  eo_init<<<512, 256, 0, stream>>>(grid, max_slot, out);
  eo_scatter_fixed<<<(NTER + 255) / 256, 256, 0, stream>>>(pos, nsx, nsy, nsz, grid);
  eo_scatter_mov<<<(NMOV + NFIL + 255) / 256, 256, 0, stream>>>(pos, nsx, nsy, nsz, grid);
  eo_reduce<<<128, 256, 0, stream>>>(grid, out, max_slot);   // 1024 waves
  eo_finalize<<<1, 64, 0, stream>>>(max_slot, out);
}